// CLAHE_56298431316298
// MI455X (gfx1250) — compile-verified
//
#include <hip/hip_runtime.h>
#include <stdint.h>

#define KW   31
#define PADW 15
#define TILE 128
#define HALO 158          // TILE + KW - 1
#define HP   130          // hsum LDS pitch (bank-conflict free for lane=row walks)
#define IMW  1024
#define IMH  1024

typedef unsigned int u32x4 __attribute__((ext_vector_type(4)));
typedef int          i32x4 __attribute__((ext_vector_type(4)));
typedef int          i32x8 __attribute__((ext_vector_type(8)));

__device__ __forceinline__ uint32_t lds_byte_offset(const void* p) {
    // flat LDS address: [63:32] = shared aperture, [31:0] = LDS byte offset
    return (uint32_t)(uintptr_t)p;
}

__global__ __launch_bounds__(256) void lcn_clahe_kernel(const float* __restrict__ x,
                                                        float* __restrict__ out) {
    __shared__ float xs [HALO * HALO];   //  99,856 B : halo tile of x
    __shared__ float hx [HALO * HP];     //  82,160 B : horizontal 31-sums of x
    __shared__ float hxx[HALO * HP];     //  82,160 B : horizontal 31-sums of x^2
                                         // total 258 KB < 320 KB WGP LDS (CDNA5)

    const int tid = threadIdx.x;
    const int tx = blockIdx.x, ty = blockIdx.y, b = blockIdx.z;
    const int ox = tx * TILE, oy = ty * TILE;       // output tile origin
    const int x0 = ox - PADW, y0 = oy - PADW;       // halo origin (may be <0)
    const float* img = x + (size_t)b * IMW * IMH;

    const bool interior = (x0 >= 0) && (y0 >= 0) &&
                          (x0 + HALO <= IMW) && (y0 + HALO <= IMH);

    if (interior) {
        // ---- Tensor Data Mover: one DMA stages the 158x158 strided tile ----
        if (tid < 32) {  // wave 0 only: TDM ignores EXEC, issues once per wave
            uint64_t ga = (uint64_t)(uintptr_t)(img + (size_t)y0 * IMW + x0);
            u32x4 g0;
            g0[0] = 1u;                                   // count = 1 (valid D#)
            g0[1] = lds_byte_offset(xs);                  // lds_addr
            g0[2] = (uint32_t)ga;                         // global_addr[31:0]
            g0[3] = (uint32_t)((ga >> 32) & 0x1FFFFFFull) // global_addr[56:32]
                    | (2u << 30);                         // type = 2 ("image")
            i32x8 g1;
            const uint32_t td0 = IMW, td1 = IMH;          // tensor dims
            const uint32_t tl0 = HALO, tl1 = HALO;        // tile dims
            const uint64_t s0  = IMW;                     // dim0 stride (elems)
            const uint64_t s1  = (uint64_t)IMW * IMH;     // dim1 stride (unused, 2D tile)
            g1[0] = (int)(2u << 16);                      // data_size = 4B; no pad/iter/mask
            g1[1] = (int)((td0 & 0xFFFFu) << 16);         // tensor_dim0[15:0]
            g1[2] = (int)((td0 >> 16) | ((td1 & 0xFFFFu) << 16));
            g1[3] = (int)((td1 >> 16) | (tl0 << 16));     // tile_dim0
            g1[4] = (int)tl1;                             // tile_dim1; tile_dim2 = 0
            g1[5] = (int)(uint32_t)s0;                    // tensor_dim0_stride[31:0]
            g1[6] = (int)((uint32_t)(s0 >> 32) | ((uint32_t)(s1 & 0xFFFFull) << 16));
            g1[7] = (int)(uint32_t)(s1 >> 16);
            i32x4 z4 = {0, 0, 0, 0};                      // groups 2/3: 2D tensor
            i32x8 z8 = {0, 0, 0, 0, 0, 0, 0, 0};
            __builtin_amdgcn_tensor_load_to_lds(g0, g1, z4, z4, z8, 0);
            __builtin_amdgcn_s_wait_tensorcnt(0);
        }
    } else {
        // ---- border tiles: guarded loads implement the zero padding ----
        for (int i = tid; i < HALO * HALO; i += 256) {
            int ly = i / HALO, lx = i - ly * HALO;
            int gy = y0 + ly, gx = x0 + lx;
            float v = 0.0f;
            if (gy >= 0 && gy < IMH && gx >= 0 && gx < IMW) v = img[gy * IMW + gx];
            xs[i] = v;
        }
    }
    __syncthreads();

    // ---- stage A: horizontal sliding 31-sums of x and x^2, one thread/row ----
    if (tid < HALO) {
        const float* row = &xs[tid * HALO];
        float sx = 0.0f, sxx = 0.0f;
        #pragma unroll
        for (int j = 0; j < KW; ++j) { float v = row[j]; sx += v; sxx += v * v; }
        float* hr  = &hx [tid * HP];
        float* hr2 = &hxx[tid * HP];
        hr[0] = sx; hr2[0] = sxx;
        for (int c = 1; c < TILE; ++c) {
            float vin = row[c + KW - 1], vout = row[c - 1];
            sx  += vin - vout;
            sxx += vin * vin - vout * vout;
            hr[c] = sx; hr2[c] = sxx;
        }
    }
    __syncthreads();

    // ---- stage B: vertical sliding 31-sums + fused pointwise, coalesced out ----
    {
        const int cc   = tid & (TILE - 1);   // lane -> consecutive columns
        const int half = tid >> 7;           // two halves of 64 rows
        const int r0   = half * (TILE / 2);
        float sx = 0.0f, sxx = 0.0f;
        #pragma unroll
        for (int j = 0; j < KW; ++j) {
            sx  += hx [(r0 + j) * HP + cc];
            sxx += hxx[(r0 + j) * HP + cc];
        }
        const float inv = 1.0f / 961.0f;     // (1/31)^2
        float* orow = out + (size_t)b * IMW * IMH + (size_t)(oy + r0) * IMW + ox + cc;
        for (int r = r0; r < r0 + TILE / 2; ++r) {
            if (r > r0) {
                sx  += hx [(r + KW - 1) * HP + cc] - hx [(r - 1) * HP + cc];
                sxx += hxx[(r + KW - 1) * HP + cc] - hxx[(r - 1) * HP + cc];
            }
            float xv   = xs[(r + PADW) * HALO + (cc + PADW)];
            float mean = sx * inv;
            float var  = sxx * inv - mean * mean;
            float stdv = sqrtf(fmaxf(var, 1e-5f));
            float nrm  = (xv - mean) / (stdv + 1e-5f);
            float sig  = 1.0f / (1.0f + __expf(-0.5f * nrm));
            *orow = xv * 0.2f + sig * 0.8f;
            orow += IMW;
        }
    }
}

extern "C" void kernel_launch(void* const* d_in, const int* in_sizes, int n_in,
                              void* d_out, int out_size, void* d_ws, size_t ws_size,
                              hipStream_t stream) {
    (void)n_in; (void)out_size; (void)d_ws; (void)ws_size;
    const float* x = (const float*)d_in[0];
    float* out = (float*)d_out;
    int batch = in_sizes[0] / (IMW * IMH);          // 32
    dim3 grid(IMW / TILE, IMH / TILE, batch);       // (8, 8, 32)
    lcn_clahe_kernel<<<grid, dim3(256), 0, stream>>>(x, out);
}